// GraphGenerator_69088843924091
// MI455X (gfx1250) — compile-verified
//
#include <hip/hip_runtime.h>
#include <math.h>

typedef __attribute__((ext_vector_type(2))) float v2f;
typedef __attribute__((ext_vector_type(8))) float v8f;

#define DFEAT 128
#define ROWSTRIDE 132          // dwords; 132 mod 64 = 4 -> conflict-free strided b64 reads
#define CSTRIDE 18             // dwords; conflict-free C-fragment spill
#define WAVES_PER_BLOCK 2
#define EPS_CS 1e-8f
#define THRESH 0.1f

__global__ __launch_bounds__(WAVES_PER_BLOCK * 32) void edge_wcos_wmma(
    const float* __restrict__ Lf, const float* __restrict__ Rf,
    const int* __restrict__ eidx, const float* __restrict__ W,
    float* __restrict__ out, int E, int nchunks)
{
    // W^2 table (2 heads x 128) + zero pad at dword 256 for B columns >= 2
    __shared__ float w2lds[260];
    // per-wave staging: [wave][L/R][16 rows x ROWSTRIDE]
    __shared__ float stage[WAVES_PER_BLOCK][2][16 * ROWSTRIDE];

    const int tid = threadIdx.x;
    for (int i = tid; i < 2 * DFEAT; i += blockDim.x) {
        float w = W[i];
        w2lds[i] = w * w;
    }
    if (tid < 4) w2lds[2 * DFEAT + tid] = 0.0f;
    __syncthreads();

    const int wave = tid >> 5;
    const int lane = tid & 31;
    const int chunk = blockIdx.x * WAVES_PER_BLOCK + wave;
    if (chunk >= nchunks) return;   // wave-uniform exit; EXEC stays all-ones for WMMA

    const int m    = lane & 15;     // edge-in-chunk (A/D row; also B/D column id)
    const int half = lane >> 4;     // K-split half per WMMA f32 layout
    const int e0   = chunk << 4;

    int em = e0 + m;
    if (em >= E) em = 0;            // clamp tail; stores are masked later
    int srow = eidx[em];            // left endpoint
    int drow = eidx[E + em];        // right endpoint

    float* Ls = &stage[wave][0][0];
    float* Rs = &stage[wave][1][0];

    // Coalesced row gather: whole wave loads one 512B row at a time into LDS
    #pragma unroll 4
    for (int r = 0; r < 16; ++r) {
        int sr = __shfl(srow, r, 32);
        int dr = __shfl(drow, r, 32);
        const float4 lv = *(const float4*)(Lf + (size_t)sr * DFEAT + lane * 4);
        const float4 rv = *(const float4*)(Rf + (size_t)dr * DFEAT + lane * 4);
        *(float4*)(Ls + r * ROWSTRIDE + lane * 4) = lv;
        *(float4*)(Rs + r * ROWSTRIDE + lane * 4) = rv;
    }
    asm volatile("s_wait_dscnt 0" ::: "memory");

    v8f c_dot = {0, 0, 0, 0, 0, 0, 0, 0};
    v8f c_ll  = {0, 0, 0, 0, 0, 0, 0, 0};
    v8f c_rr  = {0, 0, 0, 0, 0, 0, 0, 0};

    const int arow = m * ROWSTRIDE + 2 * half;   // A fragment base (k window added per step)
    const int col  = m;                          // B column this lane supplies

    #pragma unroll 4
    for (int s = 0; s < 32; ++s) {
        const int k4 = 4 * s;
        const float2 la = *(const float2*)(Ls + arow + k4);
        const float2 ra = *(const float2*)(Rs + arow + k4);
        const int bidx = (col < 2) ? (col * DFEAT + 2 * half + k4) : (2 * DFEAT);
        const float2 bb = *(const float2*)(&w2lds[bidx]);

        v2f a_lr, a_ll, a_rr, b;
        a_lr[0] = la.x * ra.x;  a_lr[1] = la.y * ra.y;
        a_ll[0] = la.x * la.x;  a_ll[1] = la.y * la.y;
        a_rr[0] = ra.x * ra.x;  a_rr[1] = ra.y * ra.y;
        b[0] = bb.x;            b[1] = bb.y;

        // D[m][h] += sum_k A[m][k] * W2[h][k]   (B shared by all three accumulators)
        c_dot = __builtin_amdgcn_wmma_f32_16x16x4_f32(false, a_lr, false, b,
                                                      (short)0, c_dot, false, false);
        c_ll  = __builtin_amdgcn_wmma_f32_16x16x4_f32(false, a_ll, false, b,
                                                      (short)0, c_ll,  false, false);
        c_rr  = __builtin_amdgcn_wmma_f32_16x16x4_f32(false, a_rr, false, b,
                                                      (short)0, c_rr,  false, false);
    }

    // Spill C fragments: [3][16 rows][CSTRIDE]; vgpr j / lanes0-15 -> M=j, lanes16-31 -> M=j+8
    float* cb = Ls;                 // staging no longer needed; 3*16*18 = 864 dwords
    const int crb = half << 3;
    #pragma unroll
    for (int j = 0; j < 8; ++j) {
        const int row = (j + crb) * CSTRIDE + col;
        cb[row]                      = c_dot[j];
        cb[16 * CSTRIDE + row]       = c_ll[j];
        cb[32 * CSTRIDE + row]       = c_rr[j];
    }
    asm volatile("s_wait_dscnt 0" ::: "memory");

    if (lane < 16) {
        const int e = e0 + lane;
        if (e < E) {
            const int r = lane * CSTRIDE;
            float d0 = cb[r + 0],                 d1 = cb[r + 1];
            float l0 = cb[16 * CSTRIDE + r + 0],  l1 = cb[16 * CSTRIDE + r + 1];
            float q0 = cb[32 * CSTRIDE + r + 0],  q1 = cb[32 * CSTRIDE + r + 1];
            float c0 = d0 / (fmaxf(sqrtf(l0), EPS_CS) * fmaxf(sqrtf(q0), EPS_CS));
            float c1 = d1 / (fmaxf(sqrtf(l1), EPS_CS) * fmaxf(sqrtf(q1), EPS_CS));
            float sim = 0.5f * (c0 + c1);
            out[e] = (sim < THRESH) ? 0.0f : sim;
        }
    }
}

extern "C" void kernel_launch(void* const* d_in, const int* in_sizes, int n_in,
                              void* d_out, int out_size, void* d_ws, size_t ws_size,
                              hipStream_t stream) {
    const float* Lf = (const float*)d_in[0];
    const float* Rf = (const float*)d_in[1];
    const int*   ei = (const int*)d_in[2];
    const float* W  = (const float*)d_in[3];
    float* out = (float*)d_out;

    const int E = in_sizes[2] / 2;
    const int nchunks = (E + 15) / 16;
    const int blocks = (nchunks + WAVES_PER_BLOCK - 1) / WAVES_PER_BLOCK;

    hipLaunchKernelGGL(edge_wcos_wmma, dim3(blocks), dim3(WAVES_PER_BLOCK * 32), 0, stream,
                       Lf, Rf, ei, W, out, E, nchunks);
}